// Loss_14164802142763
// MI455X (gfx1250) — compile-verified
//
#include <hip/hip_runtime.h>
#include <math.h>

// Problem constants (from reference)
constexpr int kB  = 64;          // batch
constexpr int kT  = 1024;        // time
constexpr int kV  = 2048;        // vocab
constexpr int kNC = 256;         // n_chars
constexpr int kNN = kT - kNC;    // n_nulls = 768
constexpr int kNJ = kNC + 1;     // 257 DP columns
constexpr int kBLK = 288;        // 9 waves of 32 (257 active lanes)
constexpr int kPF  = 16;         // prefetch distance in rows (rows are 8KB apart)
#define EPSV 1e-5f

typedef __attribute__((ext_vector_type(2))) float v2f;
typedef __attribute__((ext_vector_type(8))) float v8f;

#if defined(__has_builtin)
#if __has_builtin(__builtin_amdgcn_wmma_f32_16x16x4_f32)
#define USE_WMMA_REDUCE 1
#endif
#endif

// -----------------------------------------------------------------------------
// Kernel 1: anti-diagonal wavefront DP. One workgroup per batch.
// Lane j owns DP column j; its own previous value ("up") lives in a register,
// only the (j-1) neighbor value crosses lanes via double-buffered LDS
// (=> exactly one s_barrier per diagonal). Diagonal d reads probs row t=d-1:
// lane j's gather address is rowbase + t*V + target[b][j-1] — a fixed-index
// stream with 8KB stride, covered by global_prefetch_b8 kPF rows ahead plus a
// depth-1 register pipeline.
// -----------------------------------------------------------------------------
__global__ __launch_bounds__(kBLK) void wavefront_dp(
    const float* __restrict__ probs,
    const int*   __restrict__ target,
    const int*   __restrict__ target_lengths,
    float*       __restrict__ picked)   // [kB] scratch output
{
  __shared__ float Gbuf[2][kNJ];

  const int b = blockIdx.x;
  const int j = threadIdx.x;
  const float* rowbase = probs + (size_t)b * kT * kV;

  int tgt = 0;                                 // lane 0 / padding lanes -> phi slot
  if (j >= 1 && j <= kNC) tgt = target[b * kNC + (j - 1)];
  const int L = target_lengths[b];             // picked column, in [0,256]

  if (j < kNJ) Gbuf[0][j] = 0.0f;              // diagonal d=0: F[0][0]=0
  float g = 0.0f;                              // lane-local "up" value F[i-1][j]

  // depth-1 pipeline: preload row t=0
  float phi_n = rowbase[0];
  float pv_n  = rowbase[tgt];
  __syncthreads();

  int cur = 0;
  for (int d = 1; d <= kT; ++d) {
    const float ph = phi_n;                    // probs[b, d-1, 0]
    const float pv = pv_n;                     // probs[b, d-1, tgt_j]

    // issue loads for the next diagonal's row (t = d)
    if (d < kT) {
      const float* row = rowbase + (size_t)d * kV;
      phi_n = row[0];
      pv_n  = row[tgt];
    }
    // pull the line for row t = d+kPF into cache (gfx1250 global_prefetch_b8)
    if (d + kPF < kT) {
      __builtin_prefetch((const void*)(rowbase + (size_t)(d + kPF) * kV + tgt), 0, 1);
    }

    // neighbor value F[i][j-1] from previous diagonal
    const float gl = Gbuf[cur][(j >= 1 && j < kNJ) ? (j - 1) : 0];

    if (j < kNJ && j <= d && d <= j + kNN) {
      if (d == j) {
        // i = 0 boundary: col0 cumulative sum of log(diag0 + eps)
        g = gl + __logf(pv + EPSV);
      } else if (j == 0) {
        // j = 0 boundary: accumulate log(phi + eps)
        g = g + __logf(ph + EPSV);
      } else {
        // interior: F[i][j] = F[i][j-1] + F[i-1][j] + log(phi)+log(char)
        g = g + gl + __logf(ph + EPSV) + __logf(pv + EPSV);
      }
      // diagonal d = j + 768 carries F[768][j]: the final column
      if (d == j + kNN && j == L) picked[b] = g;
    }

    if (j < kNJ) Gbuf[cur ^ 1][j] = g;         // publish to the other buffer
    __syncthreads();                           // single barrier per diagonal
    cur ^= 1;
  }
}

// -----------------------------------------------------------------------------
// Kernel 2: -mean over the 64 picked values. The sum is a ones-matmul:
// A(16x4) holds the 64 values, B(4x16) = 1 => D[m][n] = rowsum_m via
// v_wmma_f32_16x16x4_f32 (f32-precise), folded with one shfl_xor.
// One wave, no divergence before the WMMA (EXEC must be all ones).
// -----------------------------------------------------------------------------
__global__ __launch_bounds__(32) void reduce_mean(
    const float* __restrict__ picked, float* __restrict__ out)
{
  const int lane = threadIdx.x;
#ifdef USE_WMMA_REDUCE
  // 32-bit A-matrix 16x4 layout: lanes 0-15 hold K=0..1, lanes 16-31 hold K=2..3
  const int m  = lane & 15;
  const int kb = (lane < 16) ? 0 : 2;
  v2f a;
  a.x = picked[m * 4 + kb + 0];
  a.y = picked[m * 4 + kb + 1];
  v2f ones; ones.x = 1.0f; ones.y = 1.0f;
  v8f c = {};
  v8f d = __builtin_amdgcn_wmma_f32_16x16x4_f32(
      /*neg_a=*/false, a, /*neg_b=*/false, ones,
      /*c_mod=*/(short)0, c, /*reuse_a=*/false, /*reuse_b=*/false);
  // lane L, vgpr r -> D[(L<16 ? r : r+8)][L%16]; all columns identical, so
  // per-lane 8-sum gives sum over M=0..7 (lanes 0-15) / M=8..15 (lanes 16-31)
  float s = d[0] + d[1] + d[2] + d[3] + d[4] + d[5] + d[6] + d[7];
  s += __shfl_xor(s, 16, 32);
  if (lane == 0) out[0] = -s * (1.0f / (float)kB);
#else
  float s = picked[lane] + picked[lane + 32];
  for (int off = 16; off >= 1; off >>= 1) s += __shfl_xor(s, off, 32);
  if (lane == 0) out[0] = -s * (1.0f / (float)kB);
#endif
}

extern "C" void kernel_launch(void* const* d_in, const int* in_sizes, int n_in,
                              void* d_out, int out_size, void* d_ws, size_t ws_size,
                              hipStream_t stream) {
  const float* probs          = (const float*)d_in[0];
  const int*   target         = (const int*)d_in[1];
  const int*   target_lengths = (const int*)d_in[2];
  float*       out            = (float*)d_out;
  float*       picked         = (float*)d_ws;   // 64 floats of scratch

  wavefront_dp<<<kB, kBLK, 0, stream>>>(probs, target, target_lengths, picked);
  reduce_mean<<<1, 32, 0, stream>>>(picked, out);
}